// GNNPolicy_49916109914654
// MI455X (gfx1250) — compile-verified
//
#include <hip/hip_runtime.h>
#include <math.h>

// ---------------------------------------------------------------------------
// MI455X (gfx1250) GNNPolicy forward.
//
// Exact algebraic simplifications:
//  * LN over a size-1 axis == its bias  -> ew is a scalar constant; initial
//    cons embedding is one constant 32-vector (cons features & edge_attr
//    never read).
//  * lin_rel(segsum(ew*x[src])) == ew*segsum((x@Wr^T)[src]) + br
//    -> project per-node (100K rows) not per-edge (2M), 4x less edge traffic.
//  * conv1 cons->var aggregation == deg(v) * const vector (degree count).
//
// Dense projections: V_WMMA_F32_16X16X4_F32, one wave per 16-row tile,
// [Wr | Wt] packed into one Kx16 B tile staged in LDS (branch-free fragments).
// Node buffers padded to 16-row multiples so WMMA loads/stores are unguarded.
// ---------------------------------------------------------------------------

typedef float v2f __attribute__((ext_vector_type(2)));
typedef float v8f __attribute__((ext_vector_type(8)));

__device__ __forceinline__ v8f wmma4(v2f a, v2f b, v8f c) {
  // (neg_a, A, neg_b, B, c_mod, C, reuse_a, reuse_b)
  return __builtin_amdgcn_wmma_f32_16x16x4_f32(false, a, false, b, (short)0, c,
                                               false, false);
}

// B[k][n] for packed [Wr | Wt] weight tile (each d_out x Kd, row-major).
__device__ __forceinline__ float bpick(int k, int n, const float* Wr,
                                       const float* Wt, int dout, int Kd) {
  if (k >= Kd) return 0.0f;
  if (n < 8) return (n < dout) ? Wr[n * Kd + k] : 0.0f;
  int n2 = n - 8;
  return (n2 < dout) ? Wt[n2 * Kd + k] : 0.0f;
}

// --------------------------- param prep (once) -----------------------------
// derived: [0]=ew, [1..32]=cons0, [33..40]=consRelC, [41..48]=consRootC
__global__ __launch_bounds__(64) void prep_kernel(
    const float* __restrict__ consLnB, const float* __restrict__ Wc,
    const float* __restrict__ bc, const float* __restrict__ edgeLnB,
    const float* __restrict__ Wr1, const float* __restrict__ Wt1,
    float* __restrict__ derived) {
  __shared__ float c0[32];
  int t = threadIdx.x;
  if (t < 32) {
    float v = fmaxf(Wc[t] * consLnB[0] + bc[t], 0.0f);
    c0[t] = v;
    derived[1 + t] = v;
  }
  __syncthreads();
  if (t < 8) {
    float sr = 0.f, st = 0.f;
    for (int k = 0; k < 32; ++k) {
      sr += Wr1[t * 32 + k] * c0[k];
      st += Wt1[t * 32 + k] * c0[k];
    }
    derived[33 + t] = sr;
    derived[41 + t] = st;
  }
  if (t == 0) derived[0] = edgeLnB[0];
}

// ------------------------ per-graph bounds branch --------------------------
__global__ __launch_bounds__(32) void bb_kernel(
    const float* __restrict__ bounds, const float* __restrict__ blnW,
    const float* __restrict__ blnB, const float* __restrict__ Wb,
    const float* __restrict__ bbb, float* __restrict__ bb,
    float* __restrict__ sums) {
  int t = threadIdx.x;
  if (t < 8) sums[t] = 0.0f;
  if (t < 4) {
    float x0 = bounds[0], x1 = bounds[1];
    float m = 0.5f * (x0 + x1);
    float d0 = x0 - m, d1 = x1 - m;
    float v = 0.5f * (d0 * d0 + d1 * d1);
    float rs = rsqrtf(v + 1e-5f);
    float n0 = d0 * rs * blnW[0] + blnB[0];
    float n1 = d1 * rs * blnW[1] + blnB[1];
    bb[t] = fmaxf(Wb[t * 2] * n0 + Wb[t * 2 + 1] * n1 + bbb[t], 0.0f);
  }
}

// ------------------------------ zero scratch -------------------------------
__global__ __launch_bounds__(256) void zero_kernel(float* __restrict__ p,
                                                   size_t n) {
  for (size_t i = (size_t)blockIdx.x * blockDim.x + threadIdx.x; i < n;
       i += (size_t)gridDim.x * blockDim.x)
    p[i] = 0.0f;
}

// ------------------- var embedding: LN(6) -> 32, WMMA ----------------------
__global__ __launch_bounds__(256) void embed_var_kernel(
    const float* __restrict__ X6, const float* __restrict__ lnw,
    const float* __restrict__ lnb, const float* __restrict__ W /*32x6*/,
    const float* __restrict__ bias, float* __restrict__ out32 /*padded rows*/,
    int M, int Mtiles) {
  __shared__ float Bs[8][32];       // B[k][n], K padded 6->8, N=32
  __shared__ float bs[32];
  __shared__ float As[8][16][8];
  // block-wide weight staging (before any wave-uniform early return)
  for (int i = threadIdx.x; i < 8 * 32; i += 256) {
    int k = i >> 5, n = i & 31;
    Bs[k][n] = (k < 6) ? W[n * 6 + k] : 0.0f;
  }
  if (threadIdx.x < 32) bs[threadIdx.x] = bias[threadIdx.x];
  __syncthreads();

  int wave = threadIdx.x >> 5;
  int lane = threadIdx.x & 31;
  int tile = blockIdx.x * 8 + wave;
  if (tile >= Mtiles) return;  // wave-uniform
  int row0 = tile * 16;

  if (lane < 16) {
    int r = row0 + lane;
    float x[6];
    float m = 0.f;
    if (r < M) {
      const float* p = X6 + (size_t)r * 6;
      for (int d = 0; d < 6; ++d) { x[d] = p[d]; m += x[d]; }
    } else {
      for (int d = 0; d < 6; ++d) x[d] = 0.f;
    }
    m *= (1.0f / 6.0f);
    float v = 0.f;
    for (int d = 0; d < 6; ++d) { float dd = x[d] - m; v += dd * dd; }
    v *= (1.0f / 6.0f);
    float rs = rsqrtf(v + 1e-5f);
    for (int d = 0; d < 6; ++d)
      As[wave][lane][d] = (x[d] - m) * rs * lnw[d] + lnb[d];
    As[wave][lane][6] = 0.f;
    As[wave][lane][7] = 0.f;
  }
  __builtin_amdgcn_wave_barrier();

  int half = lane >> 4, idx = lane & 15;
  for (int nt = 0; nt < 2; ++nt) {
    v8f acc = {};
    for (int ks = 0; ks < 2; ++ks) {
      int kb = ks * 4 + 2 * half;
      v2f a = {As[wave][idx][kb], As[wave][idx][kb + 1]};
      int n = nt * 16 + idx;
      v2f b = {Bs[kb][n], Bs[kb + 1][n]};
      acc = wmma4(a, b, acc);
    }
    int n = nt * 16 + idx;
    float bn = bs[n];
    for (int j = 0; j < 8; ++j) {
      int r = row0 + j + 8 * half;
      out32[(size_t)r * 32 + n] = fmaxf(acc[j] + bn, 0.0f);  // padded rows OK
    }
  }
}

// -------- node projection: OUT[r,0..15] = X[r,:] @ [Wr^T | Wt^T] (WMMA) ----
// X and OUT row counts padded to Mtiles*16 -> no guards anywhere.
__global__ __launch_bounds__(256) void proj_kernel(
    const float* __restrict__ X, int ldx, int Ksteps, int Kd,
    const float* __restrict__ Wr, const float* __restrict__ Wt, int dout,
    float* __restrict__ OUT, int Mtiles) {
  __shared__ float Bs[32][16];  // B[k][n] packed, K <= 32
  __shared__ float As[8][16][32];
  int K = Ksteps * 4;
  for (int i = threadIdx.x; i < K * 16; i += 256) {
    int k = i >> 4, n = i & 15;
    Bs[k][n] = bpick(k, n, Wr, Wt, dout, Kd);
  }
  __syncthreads();

  int wave = threadIdx.x >> 5;
  int lane = threadIdx.x & 31;
  int tile = blockIdx.x * 8 + wave;
  if (tile >= Mtiles) return;  // wave-uniform
  int row0 = tile * 16;

  for (int i = lane; i < 16 * K; i += 32) {
    int r = i / K, k = i - r * K;
    As[wave][r][k] = X[(size_t)(row0 + r) * ldx + k];
  }
  __builtin_amdgcn_wave_barrier();

  int half = lane >> 4, idx = lane & 15;
  v8f acc = {};
  for (int ks = 0; ks < Ksteps; ++ks) {
    int kb = ks * 4 + 2 * half;
    v2f a = {As[wave][idx][kb], As[wave][idx][kb + 1]};
    v2f b = {Bs[kb][idx], Bs[kb + 1][idx]};
    acc = wmma4(a, b, acc);
  }
  for (int j = 0; j < 8; ++j)
    OUT[(size_t)(row0 + j + 8 * half) * 16 + idx] = acc[j];
}

// ------------------------------ edge passes --------------------------------
__global__ __launch_bounds__(256) void edge_conv1_kernel(
    const int* __restrict__ eidx, int E, const float* __restrict__ vproj,
    float* __restrict__ aggC, float* __restrict__ deg) {
  long long tid = (long long)blockIdx.x * blockDim.x + threadIdx.x;
  long long e = tid >> 3;
  int j = (int)(tid & 7);
  if (e >= E) return;
  int v = eidx[e];
  int c = eidx[(size_t)E + e];
  atomicAdd(&aggC[(size_t)c * 8 + j], vproj[(size_t)v * 16 + j]);
  if (j == 0) atomicAdd(&deg[v], 1.0f);
}

__global__ __launch_bounds__(256) void edge_conv23_kernel(
    const int* __restrict__ eidx, int E, const float* __restrict__ vproj,
    const float* __restrict__ cproj, float* __restrict__ aggC,
    float* __restrict__ aggV) {
  long long tid = (long long)blockIdx.x * blockDim.x + threadIdx.x;
  long long e = tid >> 3;
  int j = (int)(tid & 7);
  if (e >= E) return;
  int v = eidx[e];
  int c = eidx[(size_t)E + e];
  if (j < 4) {
    atomicAdd(&aggC[(size_t)c * 8 + j], vproj[(size_t)v * 16 + j]);
  } else {
    int j2 = j - 4;
    atomicAdd(&aggV[(size_t)v * 8 + j2], cproj[(size_t)c * 16 + j2]);
  }
}

// ------------------------------- combines ----------------------------------
__global__ __launch_bounds__(256) void combine_cons1_kernel(
    const float* __restrict__ aggC, const float* __restrict__ derived,
    const float* __restrict__ br, float* __restrict__ consF, int M) {
  size_t i = (size_t)blockIdx.x * blockDim.x + threadIdx.x;
  if (i >= (size_t)M * 8) return;
  int j = (int)(i & 7);
  float ew = derived[0];
  consF[i] = fmaxf(ew * aggC[i] + br[j] + derived[41 + j], 0.0f);
}

__global__ __launch_bounds__(256) void combine_var1_kernel(
    const float* __restrict__ deg, const float* __restrict__ vproj,
    const float* __restrict__ derived, const float* __restrict__ br,
    float* __restrict__ varF, int M) {
  size_t i = (size_t)blockIdx.x * blockDim.x + threadIdx.x;
  if (i >= (size_t)M * 8) return;
  size_t r = i >> 3;
  int j = (int)(i & 7);
  float ew = derived[0];
  float v = ew * deg[r] * derived[33 + j] + br[j] + vproj[r * 16 + 8 + j];
  varF[i] = fmaxf(v, 0.0f);
}

__global__ __launch_bounds__(256) void combine_generic_kernel(
    const float* __restrict__ agg, const float* __restrict__ proj,
    const float* __restrict__ derived, const float* __restrict__ br, int dout,
    float* __restrict__ outF, int M) {
  size_t i = (size_t)blockIdx.x * blockDim.x + threadIdx.x;
  if (i >= (size_t)M * 8) return;
  size_t r = i >> 3;
  int j = (int)(i & 7);
  float v = 0.0f;
  if (j < dout)
    v = fmaxf(derived[0] * agg[i] + br[j] + proj[r * 16 + 8 + j], 0.0f);
  outF[i] = v;  // zero-pad cols >= dout so next K=8 projection is exact
}

// --------------------------- column-sum reduce -----------------------------
__global__ __launch_bounds__(256) void reduce4_kernel(
    const float* __restrict__ F, int M, float* __restrict__ sums) {
  __shared__ float sm[4][256];
  float a0 = 0.f, a1 = 0.f, a2 = 0.f, a3 = 0.f;
  for (int r = blockIdx.x * blockDim.x + threadIdx.x; r < M;
       r += gridDim.x * blockDim.x) {
    const float* p = F + (size_t)r * 8;
    a0 += p[0]; a1 += p[1]; a2 += p[2]; a3 += p[3];
  }
  sm[0][threadIdx.x] = a0; sm[1][threadIdx.x] = a1;
  sm[2][threadIdx.x] = a2; sm[3][threadIdx.x] = a3;
  __syncthreads();
  for (int s = 128; s > 0; s >>= 1) {
    if (threadIdx.x < s)
      for (int j = 0; j < 4; ++j)
        sm[j][threadIdx.x] += sm[j][threadIdx.x + s];
    __syncthreads();
  }
  if (threadIdx.x < 4) atomicAdd(&sums[threadIdx.x], sm[threadIdx.x][0]);
}

__global__ __launch_bounds__(32) void score_kernel(
    const float* __restrict__ sums, const float* __restrict__ bb, float invNV,
    float invNC, float* __restrict__ scoreSlot) {
  if (threadIdx.x == 0) {
    float s = 0.f;
    for (int j = 0; j < 4; ++j) { float a = sums[j] * invNV; s += a * a; }
    for (int j = 0; j < 4; ++j) { float a = sums[4 + j] * invNC; s += a * a; }
    for (int j = 0; j < 4; ++j) s += bb[j] * bb[j];
    *scoreSlot = sqrtf(s);
  }
}

__global__ __launch_bounds__(32) void final_kernel(
    const float* __restrict__ scores, float* __restrict__ out) {
  if (threadIdx.x == 0)
    out[0] = 1.0f / (1.0f + expf(scores[0] - scores[1]));
}

// ------------------------------- host side ---------------------------------
extern "C" void kernel_launch(void* const* d_in, const int* in_sizes, int n_in,
                              void* d_out, int out_size, void* d_ws,
                              size_t ws_size, hipStream_t stream) {
  const int NC = in_sizes[0];
  const int E = in_sizes[1] / 2;
  const int NV = in_sizes[3] / 6;

  const int* eidx[2] = {(const int*)d_in[1], (const int*)d_in[6]};
  const float* var_in[2] = {(const float*)d_in[3], (const float*)d_in[8]};
  const float* bounds[2] = {(const float*)d_in[4], (const float*)d_in[9]};
  // cons features & edge_attr are provably unused (LN over size-1 axis).

  // params: jax pytree dict-sorted leaf order
  const float* Wb = (const float*)d_in[10];
  const float* bbLb = (const float*)d_in[11];
  const float* blnW = (const float*)d_in[12];
  const float* blnB = (const float*)d_in[13];
  const float* Wc = (const float*)d_in[14];
  const float* bc = (const float*)d_in[15];
  const float* clnB = (const float*)d_in[17];
  const float* Wr1 = (const float*)d_in[18];
  const float* br1 = (const float*)d_in[19];
  const float* Wt1 = (const float*)d_in[20];
  const float* Wr2 = (const float*)d_in[21];
  const float* br2 = (const float*)d_in[22];
  const float* Wt2 = (const float*)d_in[23];
  const float* Wr3 = (const float*)d_in[24];
  const float* br3 = (const float*)d_in[25];
  const float* Wt3 = (const float*)d_in[26];
  const float* elnB = (const float*)d_in[28];
  const float* Wv = (const float*)d_in[29];
  const float* bv = (const float*)d_in[30];
  const float* vlnW = (const float*)d_in[31];
  const float* vlnB = (const float*)d_in[32];

  int tilesV = (NV + 15) / 16, tilesC = (NC + 15) / 16;
  size_t rowsV = (size_t)tilesV * 16, rowsC = (size_t)tilesC * 16;

  // workspace partition (floats); node buffers padded to 16-row multiples
  float* f = (float*)d_ws;
  size_t o = 0;
  float* var32 = f + o; o += rowsV * 32;
  float* varF = f + o;  o += rowsV * 8;
  float* consF = f + o; o += rowsC * 8;
  float* vproj = f + o; o += rowsV * 16;
  float* cproj = f + o; o += rowsC * 16;
  float* aggC = f + o;  o += (size_t)NC * 8;  // contiguous with aggV, deg
  float* aggV = f + o;  o += (size_t)NV * 8;
  float* deg = f + o;   o += (size_t)NV;
  size_t zlen = (size_t)NC * 8 + (size_t)NV * 8 + (size_t)NV;
  float* derived = f + o; o += 64;
  float* bbBuf = f + o;   o += 4;
  float* sums = f + o;    o += 8;
  float* scores = f + o;  o += 2;
  if (ws_size < o * sizeof(float)) return;  // insufficient scratch

  const int B = 256;
  int gTileV = (tilesV + 7) / 8, gTileC = (tilesC + 7) / 8;
  long long edgeThreads = (long long)E * 8;
  int gEdge = (int)((edgeThreads + B - 1) / B);
  int gNV8 = (int)(((size_t)NV * 8 + B - 1) / B);
  int gNC8 = (int)(((size_t)NC * 8 + B - 1) / B);
  int gZero = (int)((zlen + B - 1) / B);
  if (gZero > 4096) gZero = 4096;

  prep_kernel<<<1, 64, 0, stream>>>(clnB, Wc, bc, elnB, Wr1, Wt1, derived);

  for (int g = 0; g < 2; ++g) {
    bb_kernel<<<1, 32, 0, stream>>>(bounds[g], blnW, blnB, Wb, bbLb, bbBuf,
                                    sums);
    embed_var_kernel<<<gTileV, B, 0, stream>>>(var_in[g], vlnW, vlnB, Wv, bv,
                                               var32, NV, tilesV);
    // ---- conv1: 32 -> 8 ----
    zero_kernel<<<gZero, B, 0, stream>>>(aggC, zlen);
    proj_kernel<<<gTileV, B, 0, stream>>>(var32, 32, 8, 32, Wr1, Wt1, 8, vproj,
                                          tilesV);
    edge_conv1_kernel<<<gEdge, B, 0, stream>>>(eidx[g], E, vproj, aggC, deg);
    combine_cons1_kernel<<<gNC8, B, 0, stream>>>(aggC, derived, br1, consF, NC);
    combine_var1_kernel<<<gNV8, B, 0, stream>>>(deg, vproj, derived, br1, varF,
                                                NV);
    // ---- conv2: 8 -> 4 ----
    zero_kernel<<<gZero, B, 0, stream>>>(aggC, zlen);
    proj_kernel<<<gTileV, B, 0, stream>>>(varF, 8, 2, 8, Wr2, Wt2, 4, vproj,
                                          tilesV);
    proj_kernel<<<gTileC, B, 0, stream>>>(consF, 8, 2, 8, Wr2, Wt2, 4, cproj,
                                          tilesC);
    edge_conv23_kernel<<<gEdge, B, 0, stream>>>(eidx[g], E, vproj, cproj, aggC,
                                                aggV);
    combine_generic_kernel<<<gNC8, B, 0, stream>>>(aggC, cproj, derived, br2, 4,
                                                   consF, NC);
    combine_generic_kernel<<<gNV8, B, 0, stream>>>(aggV, vproj, derived, br2, 4,
                                                   varF, NV);
    // ---- conv3: 4 -> 4 (features zero-padded to K=8) ----
    zero_kernel<<<gZero, B, 0, stream>>>(aggC, zlen);
    proj_kernel<<<gTileV, B, 0, stream>>>(varF, 8, 2, 4, Wr3, Wt3, 4, vproj,
                                          tilesV);
    proj_kernel<<<gTileC, B, 0, stream>>>(consF, 8, 2, 4, Wr3, Wt3, 4, cproj,
                                          tilesC);
    edge_conv23_kernel<<<gEdge, B, 0, stream>>>(eidx[g], E, vproj, cproj, aggC,
                                                aggV);
    combine_generic_kernel<<<gNC8, B, 0, stream>>>(aggC, cproj, derived, br3, 4,
                                                   consF, NC);
    combine_generic_kernel<<<gNV8, B, 0, stream>>>(aggV, vproj, derived, br3, 4,
                                                   varF, NV);
    // ---- readout ----
    reduce4_kernel<<<512, B, 0, stream>>>(varF, NV, sums);
    reduce4_kernel<<<512, B, 0, stream>>>(consF, NC, sums + 4);
    score_kernel<<<1, 32, 0, stream>>>(sums, bbBuf, 1.0f / (float)NV,
                                       1.0f / (float)NC, scores + g);
  }
  final_kernel<<<1, 32, 0, stream>>>(scores, (float*)d_out);
}